// Quadratic_19456201851547
// MI455X (gfx1250) — compile-verified
//
#include <hip/hip_runtime.h>

// CDNA5 (gfx1250) wave32 WMMA kernel: out = b + W@X.T + diag(X Q X.T)
// f32 precision via V_WMMA_F32_16X16X4_F32.
// Q staged TRANSPOSED in LDS so each B fragment is one ds_load_b64.

typedef __attribute__((ext_vector_type(2))) float v2f;
typedef __attribute__((ext_vector_type(8))) float v8f;

#define D 128
#define QT_STRIDE 132  // Qt row stride: 132 % 64 == 4 -> conflict-free b64 B frags
#define XSTRIDE   132  // X tile row stride: conflict-free A frags + C-layout reads
#define WAVES_PER_BLOCK 4
#define TILES_PER_WAVE  4
#define ROWS_PER_BLOCK (WAVES_PER_BLOCK * TILES_PER_WAVE * 16)  // 256

static constexpr int QELEMS = D * QT_STRIDE;                 // 16896 floats
static constexpr int WOFF   = QELEMS;
static constexpr int XOFF   = WOFF + D;
static constexpr int XTILE  = 16 * XSTRIDE;                  // 2112 floats/wave
static constexpr int SMEM_FLOATS = XOFF + WAVES_PER_BLOCK * XTILE;  // ~102 KB

__global__ __launch_bounds__(128)
void quad_wmma_kernel(const float* __restrict__ X,
                      const float* __restrict__ bscalar,
                      const float* __restrict__ W,
                      const float* __restrict__ Q,
                      float* __restrict__ out)
{
    extern __shared__ float smem[];
    float* sQt = smem;                // Qt[e][d] = Q[d][e], row stride QT_STRIDE
    float* sW  = smem + WOFF;         // W[128]

    const int tid  = threadIdx.x;
    const int lane = tid & 31;
    const int wave = tid >> 5;
    float* sx = smem + XOFF + wave * XTILE;  // wave-private 16x128 X tile

    // ---- Stage Q TRANSPOSED into LDS: coalesced float4 global reads,
    //      4 scattered ds_store_b32 per chunk (one-time cost). ----
    for (int i = tid; i < D * (D / 4); i += 128) {
        const int d = i >> 5;   // Q row (K index)
        const int c = i & 31;   // float4 chunk within row (columns 4c..4c+3)
        const float4 q4 = ((const float4*)Q)[i];
        float* dst = sQt + (4 * c) * QT_STRIDE + d;
        dst[0 * QT_STRIDE] = q4.x;
        dst[1 * QT_STRIDE] = q4.y;
        dst[2 * QT_STRIDE] = q4.z;
        dst[3 * QT_STRIDE] = q4.w;
    }
    if (tid < D) sW[tid] = W[tid];
    __syncthreads();  // only block-wide barrier: Q/W visible to all waves

    const float bias = bscalar[0];
    const int   l16  = lane & 15;
    const bool  hi   = lane >= 16;

    for (int t = 0; t < TILES_PER_WAVE; ++t) {
        const int rowBase = blockIdx.x * ROWS_PER_BLOCK
                          + (wave * TILES_PER_WAVE + t) * 16;
        const float* Xg = X + (size_t)rowBase * D;

        // Compiler fence: keep this tile's LDS stores after prior tile's loads.
        // (HW: same-wave DS ops execute in order, so no hardware wait needed.)
        asm volatile("" ::: "memory");

        // ---- Stage wave-private 16x128 X tile (padded stride) ----
        for (int i = lane; i < 16 * (D / 4); i += 32) {
            const int row   = i >> 5;
            const int chunk = i & 31;
            ((float4*)(sx + row * XSTRIDE))[chunk] = ((const float4*)Xg)[i];
        }

        // Prefetch next tile's X (8 KB = 32 lanes x 256 B) while we compute.
        if (t + 1 < TILES_PER_WAVE) {
            const char* nxt = (const char*)(Xg + 16 * D);
            __builtin_prefetch(nxt + lane * 256, 0, 0);
        }

        // Drain DS pipe (same-wave in-order makes this store->load safe) and
        // fence the compiler before fragment reads.
        asm volatile("s_wait_dscnt 0" ::: "memory");

        float psum[8] = {0.f, 0.f, 0.f, 0.f, 0.f, 0.f, 0.f, 0.f};

        // T = Xtile @ Q + 1(x)W, 16 output columns at a time.
        for (int j = 0; j < D / 16; ++j) {
            const int col = j * 16 + l16;

            // Fold linear term: C[r][col] = W[col]
            const float wv = sW[col];
            v8f acc = { wv, wv, wv, wv, wv, wv, wv, wv };

            // ISA 32-bit layouts (16x16x4): lanes 0-15 carry K0/K1,
            // lanes 16-31 carry K2/K3, for both A and B fragments.
            #pragma unroll
            for (int kk = 0; kk < D / 4; ++kk) {
                const int k0 = kk * 4 + (hi ? 2 : 0);
                const v2f a  = *(const v2f*)(sx  + l16 * XSTRIDE   + k0); // ds_load_b64
                const v2f bf = *(const v2f*)(sQt + col * QT_STRIDE + k0); // ds_load_b64
                acc = __builtin_amdgcn_wmma_f32_16x16x4_f32(
                        /*neg_a=*/false, a, /*neg_b=*/false, bf,
                        /*c_mod=*/(short)0, acc,
                        /*reuse_a=*/false, /*reuse_b=*/false);
            }

            // Elementwise multiply with X in C/D layout, accumulate per lane.
            #pragma unroll
            for (int r = 0; r < 8; ++r) {
                const int row = hi ? (r + 8) : r;
                psum[r] += acc[r] * sx[row * XSTRIDE + col];
            }
        }

        // Reduce across the 16 lanes of each half.
        #pragma unroll
        for (int r = 0; r < 8; ++r) {
            float v = psum[r];
            v += __shfl_xor(v, 1, 32);
            v += __shfl_xor(v, 2, 32);
            v += __shfl_xor(v, 4, 32);
            v += __shfl_xor(v, 8, 32);
            psum[r] = v;
        }

        // Lane 0 stores rows 0..7, lane 16 stores rows 8..15.
        if (l16 == 0) {
            const int rb = rowBase + (hi ? 8 : 0);
            #pragma unroll
            for (int r = 0; r < 8; ++r)
                out[rb + r] = bias + psum[r];
        }
    }
}

extern "C" void kernel_launch(void* const* d_in, const int* in_sizes, int n_in,
                              void* d_out, int out_size, void* d_ws, size_t ws_size,
                              hipStream_t stream) {
    const float* X = (const float*)d_in[0];   // (N, 128) f32
    const float* b = (const float*)d_in[1];   // scalar f32
    const float* W = (const float*)d_in[2];   // (1, 128) f32
    const float* Q = (const float*)d_in[3];   // (128, 128) f32
    float* out = (float*)d_out;               // (1, N) f32

    const int N = in_sizes[0] / D;            // 262144
    dim3 grid(N / ROWS_PER_BLOCK);            // 1024
    dim3 block(128);                          // 4 wave32s
    const size_t smem_bytes = (size_t)SMEM_FLOATS * sizeof(float);

    quad_wmma_kernel<<<grid, block, smem_bytes, stream>>>(X, b, W, Q, out);
}